// ConvReverseDataNet_10737418240060
// MI455X (gfx1250) — compile-verified
//
#include <hip/hip_runtime.h>
#include <hip/hip_bf16.h>

// ---------------------------------------------------------------------------
// USRNet FFT data-step for MI455X (gfx1250), wave32.
// 256-point FFTs done as four-step (16x16) with V_WMMA_F32_16X16X4_F32.
// ---------------------------------------------------------------------------

typedef float v2f __attribute__((ext_vector_type(2)));
typedef float v8f __attribute__((ext_vector_type(8)));

#define PI_F 3.14159265358979323846f
#define LDSP 17   // padded LDS row stride (16 + 1) -> conflict-free column reads

__device__ __forceinline__ v8f wmma4(v2f a, v2f b, v8f c) {
  // D(16x16,f32) = A(16x4,f32) x B(4x16,f32) + C
  return __builtin_amdgcn_wmma_f32_16x16x4_f32(
      /*neg_a=*/false, a, /*neg_b=*/false, b,
      /*c_mod=*/(short)0, c, /*reuse_a=*/false, /*reuse_b=*/false);
}

// ---------------------------------------------------------------------------
// One 256-point complex FFT per wave, four-step algorithm (N1=N2=16).
//   D[n2][n1] = x[n1 + 16*n2]  (natural flat order in padded LDS)
//   T = F16 x D ; U = T .* w256^(M*N) ; V = U x F16
//   out[16*k1 + k2] = V[k2][k1]  -> flat idx = M + 16*N in C-layout terms
// TS = -1 forward, +1 inverse.  F16[m][n] = exp(TS*i*2*pi*m*n/16).
// ---------------------------------------------------------------------------
template<int TS>
__device__ __forceinline__ void fft256_wave(const float* Dr, const float* Di,
                                            float* Ur, float* Ui,
                                            v8f& vr, v8f& vi) {
  const int lane = (int)(threadIdx.x & 31);
  const int half = lane >> 4;   // 0 or 1
  const int l15  = lane & 15;

  // ---- Step 1: T = F16 x D  (A = F16 analytic, B = D from LDS) ----
  v8f tr = {0,0,0,0,0,0,0,0};
  v8f ti = {0,0,0,0,0,0,0,0};
#pragma unroll
  for (int k0 = 0; k0 < 16; k0 += 4) {
    const int kb = k0 + 2*half;                        // A/B K-lane split
    const float th0 = (PI_F/8.0f) * (float)((l15*kb)     & 15);
    const float th1 = (PI_F/8.0f) * (float)((l15*(kb+1)) & 15);
    v2f aC  = { __cosf(th0), __cosf(th1) };
    v2f aS  = { (float)TS*__sinf(th0), (float)TS*__sinf(th1) };
    v2f aSn = { -aS[0], -aS[1] };
    v2f bR  = { Dr[kb*LDSP + l15], Dr[(kb+1)*LDSP + l15] };
    v2f bI  = { Di[kb*LDSP + l15], Di[(kb+1)*LDSP + l15] };
    // Yr = C*Xr + (-TS*sin)*Xi ; Yi = C*Xi + (TS*sin)*Xr
    tr = wmma4(aC,  bR, tr);
    tr = wmma4(aSn, bI, tr);
    ti = wmma4(aC,  bI, ti);
    ti = wmma4(aS,  bR, ti);
  }

  // ---- Step 2: twiddle by w256^(M*N), stage U into LDS ----
#pragma unroll
  for (int v = 0; v < 8; ++v) {
    const int M = v + 8*half;
    const int N = l15;
    const float ang = (float)TS * (PI_F/128.0f) * (float)((M*N) & 255);
    const float c = __cosf(ang), s = __sinf(ang);
    const float a = tr[v], b = ti[v];
    Ur[M*LDSP + N] = a*c - b*s;
    Ui[M*LDSP + N] = a*s + b*c;
  }
  __builtin_amdgcn_wave_barrier();

  // ---- Step 3: V = U x F16  (A = U from LDS, B = F16 analytic) ----
  v8f zr = {0,0,0,0,0,0,0,0};
  vr = zr; vi = zr;
#pragma unroll
  for (int k0 = 0; k0 < 16; k0 += 4) {
    const int kb = k0 + 2*half;
    v2f aR = { Ur[l15*LDSP + kb], Ur[l15*LDSP + kb + 1] };
    v2f aI = { Ui[l15*LDSP + kb], Ui[l15*LDSP + kb + 1] };
    const float th0 = (PI_F/8.0f) * (float)((kb*l15)     & 15);
    const float th1 = (PI_F/8.0f) * (float)(((kb+1)*l15) & 15);
    v2f bC  = { __cosf(th0), __cosf(th1) };
    v2f bS  = { (float)TS*__sinf(th0), (float)TS*__sinf(th1) };
    v2f bSn = { -bS[0], -bS[1] };
    // Vr = Ur*C + Ui*(-TS*sin) ; Vi = Ur*(TS*sin) + Ui*C
    vr = wmma4(aR, bC,  vr);
    vr = wmma4(aI, bSn, vr);
    vi = wmma4(aR, bS,  vi);
    vi = wmma4(aI, bC,  vi);
  }
}

// ===========================================================================
// Pass kernels: 256 threads = 8 waves, one line per wave.
// Forward passes write transposed so the next pass reads contiguous lines.
// ===========================================================================

// Row FFT of the rolled/zero-padded PSF. outT[(plane*256 + c_f)*256 + r].
__global__ void __launch_bounds__(256) kRowPSF(const float* __restrict__ psf,
                                               float* __restrict__ outT) {
  __shared__ float sD[8][2][16*LDSP];
  __shared__ float sU[8][2][16*LDSP];
  const int wave = (int)(threadIdx.x >> 5);
  const int lane = (int)(threadIdx.x & 31);
  float* Dr = sD[wave][0]; float* Di = sD[wave][1];
  float* Ur = sU[wave][0]; float* Ui = sU[wave][1];

  const long long line  = (long long)blockIdx.x * 8 + wave;   // plane*256 + r
  const int plane = (int)(line >> 8);
  const int r     = (int)(line & 255);
  const int rr    = (r + 12) & 255;   // roll by -(25/2)=-12
#pragma unroll
  for (int v = 0; v < 8; ++v) {
    const int e  = lane + 32*v;
    const int cc = (e + 12) & 255;
    float val = 0.0f;
    if (rr < 25 && cc < 25) val = psf[((long long)plane*25 + rr)*25 + cc];
    Dr[(e>>4)*LDSP + (e&15)] = val;
    Di[(e>>4)*LDSP + (e&15)] = 0.0f;
  }
  v8f vr, vi;
  fft256_wave<-1>(Dr, Di, Ur, Ui, vr, vi);
  const int half = lane >> 4, l15 = lane & 15;
#pragma unroll
  for (int v = 0; v < 8; ++v) {
    const int idx = 16*l15 + v + 8*half;
    const long long o = (((long long)plane*256 + idx)*256 + r)*2;
    outT[o] = vr[v]; outT[o+1] = vi[v];
  }
}

// Row FFT of STy (2x zero-upsampled x). Odd rows are zero.
__global__ void __launch_bounds__(256) kRowSTy(const float* __restrict__ x,
                                               float* __restrict__ outT) {
  __shared__ float sD[8][2][16*LDSP];
  __shared__ float sU[8][2][16*LDSP];
  const int wave = (int)(threadIdx.x >> 5);
  const int lane = (int)(threadIdx.x & 31);
  float* Dr = sD[wave][0]; float* Di = sD[wave][1];
  float* Ur = sU[wave][0]; float* Ui = sU[wave][1];

  const long long line  = (long long)blockIdx.x * 8 + wave;
  const int plane = (int)(line >> 8);
  const int r     = (int)(line & 255);
#pragma unroll
  for (int v = 0; v < 8; ++v) {
    const int e = lane + 32*v;
    float val = 0.0f;
    if (((r & 1) == 0) && ((e & 1) == 0))
      val = x[((long long)plane*128 + (r >> 1))*128 + (e >> 1)];
    Dr[(e>>4)*LDSP + (e&15)] = val;
    Di[(e>>4)*LDSP + (e&15)] = 0.0f;
  }
  v8f vr, vi;
  fft256_wave<-1>(Dr, Di, Ur, Ui, vr, vi);
  const int half = lane >> 4, l15 = lane & 15;
#pragma unroll
  for (int v = 0; v < 8; ++v) {
    const int idx = 16*l15 + v + 8*half;
    const long long o = (((long long)plane*256 + idx)*256 + r)*2;
    outT[o] = vr[v]; outT[o+1] = vi[v];
  }
}

// Column FFT: reads transposed rows (contiguous), writes natural layout.
// inT[(plane*256 + w)*256 + h], out[(plane*256 + r_f)*outW + w].
__global__ void __launch_bounds__(256) kColFwd(const float* __restrict__ inT,
                                               float* __restrict__ outN,
                                               int wPerPlane, int outW) {
  __shared__ float sD[8][2][16*LDSP];
  __shared__ float sU[8][2][16*LDSP];
  const int wave = (int)(threadIdx.x >> 5);
  const int lane = (int)(threadIdx.x & 31);
  float* Dr = sD[wave][0]; float* Di = sD[wave][1];
  float* Ur = sU[wave][0]; float* Ui = sU[wave][1];

  const long long line = (long long)blockIdx.x * 8 + wave;
  const int plane = (int)(line / wPerPlane);
  const int w     = (int)(line - (long long)plane * wPerPlane);
#pragma unroll
  for (int v = 0; v < 8; ++v) {
    const int e = lane + 32*v;
    const long long o = (((long long)plane*256 + w)*256 + e)*2;
    Dr[(e>>4)*LDSP + (e&15)] = inT[o];
    Di[(e>>4)*LDSP + (e&15)] = inT[o+1];
  }
  v8f vr, vi;
  fft256_wave<-1>(Dr, Di, Ur, Ui, vr, vi);
  const int half = lane >> 4, l15 = lane & 15;
#pragma unroll
  for (int v = 0; v < 8; ++v) {
    const int idx = 16*l15 + v + 8*half;
    const long long o = (((long long)plane*256 + idx)*outW + w)*2;
    outN[o] = vr[v]; outN[o+1] = vi[v];
  }
}

// Inverse row FFT of FX (natural layout), writes transposed.
__global__ void __launch_bounds__(256) kRowInv(const float* __restrict__ inN,
                                               float* __restrict__ outT) {
  __shared__ float sD[8][2][16*LDSP];
  __shared__ float sU[8][2][16*LDSP];
  const int wave = (int)(threadIdx.x >> 5);
  const int lane = (int)(threadIdx.x & 31);
  float* Dr = sD[wave][0]; float* Di = sD[wave][1];
  float* Ur = sU[wave][0]; float* Ui = sU[wave][1];

  const long long line = (long long)blockIdx.x * 8 + wave;
  const int plane = (int)(line >> 8);
  const int r     = (int)(line & 255);
#pragma unroll
  for (int v = 0; v < 8; ++v) {
    const int e = lane + 32*v;
    const long long o = (((long long)plane*256 + r)*256 + e)*2;
    Dr[(e>>4)*LDSP + (e&15)] = inN[o];
    Di[(e>>4)*LDSP + (e&15)] = inN[o+1];
  }
  v8f vr, vi;
  fft256_wave<1>(Dr, Di, Ur, Ui, vr, vi);
  const int half = lane >> 4, l15 = lane & 15;
#pragma unroll
  for (int v = 0; v < 8; ++v) {
    const int idx = 16*l15 + v + 8*half;
    const long long o = (((long long)plane*256 + idx)*256 + r)*2;
    outT[o] = vr[v]; outT[o+1] = vi[v];
  }
}

// Inverse column FFT; emits real part / 65536 in natural layout.
__global__ void __launch_bounds__(256) kColInvReal(const float* __restrict__ inT,
                                                   float* __restrict__ outR) {
  __shared__ float sD[8][2][16*LDSP];
  __shared__ float sU[8][2][16*LDSP];
  const int wave = (int)(threadIdx.x >> 5);
  const int lane = (int)(threadIdx.x & 31);
  float* Dr = sD[wave][0]; float* Di = sD[wave][1];
  float* Ur = sU[wave][0]; float* Ui = sU[wave][1];

  const long long line = (long long)blockIdx.x * 8 + wave;
  const int plane = (int)(line >> 8);
  const int w     = (int)(line & 255);
#pragma unroll
  for (int v = 0; v < 8; ++v) {
    const int e = lane + 32*v;
    const long long o = (((long long)plane*256 + w)*256 + e)*2;
    Dr[(e>>4)*LDSP + (e&15)] = inT[o];
    Di[(e>>4)*LDSP + (e&15)] = inT[o+1];
  }
  v8f vr, vi;
  fft256_wave<1>(Dr, Di, Ur, Ui, vr, vi);
  const int half = lane >> 4, l15 = lane & 15;
#pragma unroll
  for (int v = 0; v < 8; ++v) {
    const int idx = 16*l15 + v + 8*half;
    outR[((long long)plane*256 + idx)*256 + w] = vr[v] * (1.0f/65536.0f);
  }
}

// ---------------------------------------------------------------------------
// Pointwise frequency solve. One thread per (plane, i<128, j<128).
// FSTy is 128-periodic => read once; FR_p = s*(conj(FB_p) + be*Bphase_p).
// FX at the 4 alias points written to natural-layout buffer.
// ---------------------------------------------------------------------------
__global__ void __launch_bounds__(256) kPoint(const float* __restrict__ FB,
                                              const float* __restrict__ FS,
                                              const float* __restrict__ alpha,
                                              float* __restrict__ FX) {
  const long long t = (long long)blockIdx.x * 256 + threadIdx.x;
  const int j     = (int)(t & 127);
  const int i     = (int)((t >> 7) & 127);
  const int plane = (int)(t >> 14);
  const int ch    = plane & 63;
  const float be  = 1.0f / (1.0f + __expf(9.0f - alpha[ch])) + 1e-3f;

  const long long so = (((long long)plane*256 + i)*128 + j)*2;
  const float sr = FS[so], si = FS[so+1];

  float frr[4], fri[4], fbr[4], fbi[4];
  float accR = 0.f, accI = 0.f, accW = 0.f;
#pragma unroll
  for (int d = 0; d < 4; ++d) {
    const int pr = i + 128*(d >> 1);
    const int pc = j + 128*(d & 1);
    const long long o = (((long long)plane*256 + pr)*256 + pc)*2;
    const float br = FB[o], bi = FB[o+1];
    // Bphase = (1+e^{-i*pi*pr/128})(1+e^{-i*pi*pc/128})
    const float a1 = -(PI_F/128.0f)*(float)pr;
    const float a2 = -(PI_F/128.0f)*(float)pc;
    const float p1r = 1.0f + __cosf(a1), p1i = __sinf(a1);
    const float p2r = 1.0f + __cosf(a2), p2i = __sinf(a2);
    const float bpr = p1r*p2r - p1i*p2i;
    const float bpi = p1r*p2i + p1i*p2r;
    const float gr = br + be*bpr;      // conj(FB) + be*Bphase
    const float gi = -bi + be*bpi;
    const float fr = sr*gr - si*gi;    // FR = s*G
    const float fi = sr*gi + si*gr;
    frr[d] = fr; fri[d] = fi; fbr[d] = br; fbi[d] = bi;
    accR += br*fr - bi*fi;             // FB*FR
    accI += br*fi + bi*fr;
    accW += br*br + bi*bi;             // |FB|^2
  }
  accR *= 0.25f; accI *= 0.25f;
  const float inv = 1.0f / (accW*0.25f + be);
  const float wr = accR*inv, wi = accI*inv;   // invWBR
  const float ibe = 1.0f / be;
#pragma unroll
  for (int d = 0; d < 4; ++d) {
    const int pr = i + 128*(d >> 1);
    const int pc = j + 128*(d & 1);
    const float cr = fbr[d]*wr + fbi[d]*wi;   // conj(FB)*invWBR
    const float ci = fbr[d]*wi - fbi[d]*wr;
    const long long o = (((long long)plane*256 + pr)*256 + pc)*2;
    FX[o]   = (frr[d] - cr)*ibe;
    FX[o+1] = (fri[d] - ci)*ibe;
  }
}

// ---------------------------------------------------------------------------
// d_in: x(4,64,128,128) f32 | k(4,64,25,25) f32 | alpha(64) f32 | sf(=2)
// d_out: (4,64,256,256) f32
// ws: C1 (134MB) | C2 (134MB) | C3 compact FSTy (67MB)
// ---------------------------------------------------------------------------
extern "C" void kernel_launch(void* const* d_in, const int* in_sizes, int n_in,
                              void* d_out, int out_size, void* d_ws, size_t ws_size,
                              hipStream_t stream) {
  const float* x     = (const float*)d_in[0];
  const float* psf   = (const float*)d_in[1];
  const float* alpha = (const float*)d_in[2];
  float* out = (float*)d_out;

  float* C1 = (float*)d_ws;                       // 256*65536*2 floats
  float* C2 = C1 + (long long)256*65536*2;        // 256*65536*2 floats
  float* C3 = C2 + (long long)256*65536*2;        // 256*256*128*2 floats

  const dim3 blk(256);
  kRowPSF    <<<8192,  blk, 0, stream>>>(psf, C1);          // PSF rows -> C1 (T)
  kColFwd    <<<8192,  blk, 0, stream>>>(C1, C2, 256, 256); // -> FB natural (C2)
  kRowSTy    <<<8192,  blk, 0, stream>>>(x, C1);            // STy rows -> C1 (T)
  kColFwd    <<<4096,  blk, 0, stream>>>(C1, C3, 128, 128); // -> FSTy compact (C3)
  kPoint     <<<16384, blk, 0, stream>>>(C2, C3, alpha, C1);// -> FX natural (C1)
  kRowInv    <<<8192,  blk, 0, stream>>>(C1, C2);           // IFFT rows -> C2 (T)
  kColInvReal<<<8192,  blk, 0, stream>>>(C2, out);          // IFFT cols -> real out
}